// BaseModel_46737834115483
// MI455X (gfx1250) — compile-verified
//
#include <hip/hip_runtime.h>

typedef __attribute__((ext_vector_type(2))) float v2f;
typedef __attribute__((ext_vector_type(8))) float v8f;

#define BETA 0.9f
#define THR  1.0f

#define B_    256
#define C_IN  30
#define T_    512
#define L1    24
#define L2    20
#define L3    18
#define C1    32
#define C2    64
#define C3    64
#define K2TOT 160   // 32*5
#define K3TOT 192   // 64*3
// row-pair interleaved spike rasters: elem(ci,p) at (ci>>1)*2*PITCH + p*2 + (ci&1)
#define S1COLS 40   // >= 24 + 4(k) + pad: Ntile1 reads up to col 35
#define S2COLS 36   // >= 20 + 2(k) + pad: Ntile1 reads up to col 33
#define S1RP   (2 * S1COLS)   // 80 floats per row-pair
#define S2RP   (2 * S2COLS)   // 72 floats per row-pair

__global__ __launch_bounds__(256)
void snn_scan_kernel(const float* __restrict__ x,
                     const float* __restrict__ W1, const float* __restrict__ b1,
                     const float* __restrict__ W2, const float* __restrict__ b2,
                     const float* __restrict__ W3, const float* __restrict__ b3,
                     const float* __restrict__ Wfc, const float* __restrict__ bfc,
                     float* __restrict__ out)
{
    __shared__ float xtL[32];
    __shared__ float s1P[(C1 / 2) * S1RP];   // 1280
    __shared__ float s2P[(C2 / 2) * S2RP];   // 2304
    __shared__ float W1L[C1 * 7];            // 224
    __shared__ float b1L[32], b2L[64], b3L[64];
    __shared__ float W2A[C2 * K2TOT];        // 10240: [co][k*32+ci]
    __shared__ float W3A[C3 * K3TOT];        // 12288: [co][k*64+ci]
    __shared__ float accL[C3 * L3];          // 1152

    const int tid  = threadIdx.x;
    const int b    = blockIdx.x;
    const int lane = tid & 31;
    const int wv   = tid >> 5;            // 0..7 waves
    const int mtile = wv >> 1;            // 0..3
    const int ntile = wv & 1;             // 0..1
    const int lm    = lane & 15;          // A: row m / B: col n within tile
    const int kslot = (lane >> 4) << 1;   // K sub-offset: 0 or 2
    const int p2    = ntile * 16 + lm;    // output column (position)
    const int mrow  = (lane >> 4) << 3;   // C/D: M base 0 or 8
    const bool valid2 = (p2 < L2);
    const bool valid3 = (p2 < L3);

    // ---- stage weights into LDS (rearranged K = k*ci_count + ci) ----
    for (int i = tid; i < C1 * 7; i += 256) W1L[i] = W1[i];
    if (tid < 32) b1L[tid] = b1[tid];
    if (tid < 64) { b2L[tid] = b2[tid]; b3L[tid] = b3[tid]; }
    for (int i = tid; i < C2 * K2TOT; i += 256) {
        int co = i / K2TOT, kk = i % K2TOT;
        int ci = kk & 31, k = kk >> 5;
        W2A[i] = W2[(co * 32 + ci) * 5 + k];
    }
    for (int i = tid; i < C3 * K3TOT; i += 256) {
        int co = i / K3TOT, kk = i % K3TOT;
        int ci = kk & 63, k = kk >> 6;
        W3A[i] = W3[(co * 64 + ci) * 3 + k];
    }
    __syncthreads();

    // ---- lane-constant bases (all LDS address math hoisted) ----
    const float* a2base = &W2A[(mtile * 16 + lm) * K2TOT + kslot];
    const float* a3base = &W3A[(mtile * 16 + lm) * K3TOT + kslot];
    // B source: row = ci0 + kslot (always even) -> pair {row,row+1} contiguous
    const float* b2src  = &s1P[(kslot >> 1) * S1RP + p2 * 2];
    const float* b3src  = &s2P[(kslot >> 1) * S2RP + p2 * 2];
    // s2 spike writes: co = mtile*16 + mrow + v  ->  (co>>1)*S2RP + p2*2 + (v&1)
    float* s2dst = &s2P[((mtile * 16 + mrow) >> 1) * S2RP + p2 * 2];

    // conv1 per-thread constants: 3 outputs share one channel (3 | 24)
    const int c1c = tid / 8;               // channel = (tid*3)/24
    const int p1b = (tid * 3) % L1;        // first position
    float w1r[7], b1r;
    #pragma unroll
    for (int j = 0; j < 7; ++j) w1r[j] = W1L[c1c * 7 + j];
    b1r = b1L[c1c];
    const float* xtp = &xtL[p1b];
    float* s1w = &s1P[(c1c >> 1) * S1RP + p1b * 2 + (c1c & 1)];

    // per-lane bias vectors, used as WMMA C initializer (no epilogue adds)
    v8f bias2v, bias3v;
    #pragma unroll
    for (int v = 0; v < 8; ++v) {
        bias2v[v] = b2L[mtile * 16 + mrow + v];
        bias3v[v] = b3L[mtile * 16 + mrow + v];
    }

    // ---- persistent state (registers, zero-init like reference) ----
    float m1[3] = {0.f, 0.f, 0.f};
    v8f m2 = {}; v8f m3 = {}; v8f acc = {};

    const float* xb = x + (size_t)b * C_IN * T_;

    for (int t = 0; t < T_; ++t) {
        // load x[b, :, t]  (sequential in t per channel -> cacheline reuse)
        if (tid < C_IN) xtL[tid] = xb[tid * T_ + t];
        __syncthreads();

        // ---- conv1 + LIF1 : 3 outputs per thread, weights in regs ----
        #pragma unroll
        for (int i = 0; i < 3; ++i) {
            float h = b1r;
            #pragma unroll
            for (int j = 0; j < 7; ++j) h += w1r[j] * xtp[i + j];
            float mm  = m1[i];
            float rst = (mm > THR) ? THR : 0.f;   // reset uses incoming mem
            mm = BETA * mm + h - rst;
            m1[i] = mm;
            s1w[i * 2] = (mm > THR) ? 1.f : 0.f;
        }
        __syncthreads();

        // ---- conv2 GEMM: D(64x20) = W2A(64x160) x im2col(s1)(160x32) ----
        v8f c2 = bias2v;   // bias folded into accumulator init
        #pragma unroll
        for (int s = 0; s < K2TOT / 4; ++s) {
            const int kkb = s * 4;
            const int k   = kkb >> 5;      // filter tap (compile-time)
            const int ci0 = kkb & 31;      // input-channel base (compile-time)
            v2f a  = *(const v2f*)(a2base + kkb);
            v2f bb = *(const v2f*)(b2src + (ci0 >> 1) * S1RP + k * 2);
            c2 = __builtin_amdgcn_wmma_f32_16x16x4_f32(
                    false, a, false, bb, (short)0, c2, false, false);
        }
        // LIF2 (mem2 in registers)
        #pragma unroll
        for (int v = 0; v < 8; ++v) {
            float mm  = m2[v];
            float rst = (mm > THR) ? THR : 0.f;
            mm = BETA * mm + c2[v] - rst;
            m2[v] = mm;
            if (valid2) s2dst[(v >> 1) * S2RP + (v & 1)] = (mm > THR) ? 1.f : 0.f;
        }
        __syncthreads();

        // ---- conv3 GEMM: D(64x18) = W3A(64x192) x im2col(s2)(192x32) ----
        v8f c3 = bias3v;
        #pragma unroll
        for (int s = 0; s < K3TOT / 4; ++s) {
            const int kkb = s * 4;
            const int k   = kkb >> 6;
            const int ci0 = kkb & 63;
            v2f a  = *(const v2f*)(a3base + kkb);
            v2f bb = *(const v2f*)(b3src + (ci0 >> 1) * S2RP + k * 2);
            c3 = __builtin_amdgcn_wmma_f32_16x16x4_f32(
                    false, a, false, bb, (short)0, c3, false, false);
        }
        // LIF3 + spike accumulation (all registers)
        #pragma unroll
        for (int v = 0; v < 8; ++v) {
            float mm  = m3[v];
            float rst = (mm > THR) ? THR : 0.f;
            mm = BETA * mm + c3[v] - rst;
            m3[v] = mm;
            if (valid3) acc[v] += (mm > THR) ? 1.f : 0.f;
        }
        // no barrier needed here: next iteration's two barriers already
        // separate every LDS rewrite from this iteration's reads
    }
    __syncthreads();

    // ---- writeout: out | mem1 | mem2 | mem3 (flat, return order) ----
    const size_t M1_OFF = (size_t)B_ * 128;
    const size_t M2_OFF = M1_OFF + (size_t)B_ * C1 * L1;
    const size_t M3_OFF = M2_OFF + (size_t)B_ * C2 * L2;

    #pragma unroll
    for (int i = 0; i < 3; ++i)
        out[M1_OFF + (size_t)b * (C1 * L1) + c1c * L1 + p1b + i] = m1[i];

    #pragma unroll
    for (int v = 0; v < 8; ++v) {
        int co = mtile * 16 + mrow + v;
        if (valid2) out[M2_OFF + (size_t)b * (C2 * L2) + co * L2 + p2] = m2[v];
        if (valid3) {
            out[M3_OFF + (size_t)b * (C3 * L3) + co * L3 + p2] = m3[v];
            accL[co * L3 + p2] = acc[v];
        }
    }
    __syncthreads();

    // fc: out[b][o] = bfc[o] + (1/512) * sum_j accL[j] * Wfc[o][j]
    if (tid < 128) {
        const float* wrow = Wfc + (size_t)tid * (C3 * L3);
        float sum = 0.f;
        for (int j = 0; j < C3 * L3; ++j) sum += accL[j] * wrow[j];
        out[(size_t)b * 128 + tid] = bfc[tid] + sum * (1.0f / 512.0f);
    }
}

extern "C" void kernel_launch(void* const* d_in, const int* in_sizes, int n_in,
                              void* d_out, int out_size, void* d_ws, size_t ws_size,
                              hipStream_t stream) {
    (void)in_sizes; (void)n_in; (void)out_size; (void)d_ws; (void)ws_size;
    const float* x   = (const float*)d_in[0];
    const float* W1  = (const float*)d_in[1];
    const float* b1  = (const float*)d_in[2];
    const float* W2  = (const float*)d_in[3];
    const float* b2  = (const float*)d_in[4];
    const float* W3  = (const float*)d_in[5];
    const float* b3  = (const float*)d_in[6];
    const float* Wfc = (const float*)d_in[7];
    const float* bfc = (const float*)d_in[8];
    float* out = (float*)d_out;

    snn_scan_kernel<<<B_, 256, 0, stream>>>(x, W1, b1, W2, b2, W3, b3, Wfc, bfc, out);
}